// AttentionFusion_12695923327142
// MI455X (gfx1250) — compile-verified
//
#include <hip/hip_runtime.h>
#include <math.h>
#include <stdint.h>

// Shapes fixed by the reference: B=4, S=4096, D=768 -> rows=16384, H=2D=1536.
#define D_FEAT 768
#define H_FEAT 1536
#define CH 192            // float4 chunks per 768-float half-row
#define ROWS_PER_BLK 8
#define THREADS 256
#define LN_EPS 1e-5f

typedef int v4i __attribute__((ext_vector_type(4)));
typedef __attribute__((address_space(1))) v4i* as1_v4i_p;
typedef __attribute__((address_space(3))) v4i* as3_v4i_p;

// ---- CDNA5 async global -> LDS copy (16 bytes per lane) -------------------
__device__ __forceinline__ void async_copy16(const float* __restrict__ base,
                                             unsigned byte_off,
                                             void* lds_generic) {
#if __has_builtin(__builtin_amdgcn_global_load_async_to_lds_b128)
  __builtin_amdgcn_global_load_async_to_lds_b128(
      (as1_v4i_p)(uintptr_t)((const char*)base + byte_off),
      (as3_v4i_p)(unsigned)(uintptr_t)lds_generic,
      /*imm offset*/0, /*cpol*/0);
#else
  unsigned laddr = (unsigned)(uintptr_t)lds_generic;  // low 32 bits = LDS byte address
  asm volatile("global_load_async_to_lds_b128 %0, %1, %2 offset:0"
               :: "v"(laddr), "v"(byte_off), "s"(base)
               : "memory");
#endif
}

__device__ __forceinline__ void wait_async0() {
#if __has_builtin(__builtin_amdgcn_s_wait_asynccnt)
  __builtin_amdgcn_s_wait_asynccnt(0);
#else
  asm volatile("s_wait_asynccnt 0x0" ::: "memory");
#endif
}

// ---- Main fused kernel: 1 wave32 per row, 8 rows per block ----------------
// (Defined first so the device-asm snippet shows the async staging prologue.)
__global__ __launch_bounds__(THREADS) void ln_gate_fuse(
    const float* __restrict__ seq, const float* __restrict__ msa,
    const float* __restrict__ a_vec, const float* __restrict__ scal,
    float* __restrict__ out, int nrows) {
  __shared__ __align__(16) float sA[H_FEAT];   // 6 KB staged once per block

  const int t = threadIdx.x;
  // Stage a_vec (384 x 16B chunks) into LDS via CDNA5 async copies.
  async_copy16(a_vec, (unsigned)t * 16u, &sA[t * 4]);
  if (t < (H_FEAT / 4) - 256)
    async_copy16(a_vec, (unsigned)(t + 256) * 16u, &sA[(t + 256) * 4]);
  wait_async0();
  __syncthreads();

  const int wave = t >> 5;
  const int lane = t & 31;
  const int row  = blockIdx.x * ROWS_PER_BLK + wave;
  if (row >= nrows) return;

  const float4* __restrict__ s4 = (const float4*)seq + (size_t)row * CH;
  const float4* __restrict__ m4 = (const float4*)msa + (size_t)row * CH;
  const float4* a4 = (const float4*)sA;

  float4 sv[6], mv[6];
  float s1 = 0.f, s2 = 0.f, s3 = 0.f;
#pragma unroll
  for (int j = 0; j < 6; ++j) {
    const int c = lane + 32 * j;
    const float4 x = s4[c];
    const float4 y = m4[c];
    sv[j] = x; mv[j] = y;
    const float4 ax = a4[c];
    const float4 ay = a4[CH + c];
    s1 += ((x.x + x.y) + (x.z + x.w)) + ((y.x + y.y) + (y.z + y.w));
    s2 += (x.x * x.x + x.y * x.y) + (x.z * x.z + x.w * x.w)
        + (y.x * y.x + y.y * y.y) + (y.z * y.z + y.w * y.w);
    s3 += (x.x * ax.x + x.y * ax.y) + (x.z * ax.z + x.w * ax.w)
        + (y.x * ay.x + y.y * ay.y) + (y.z * ay.z + y.w * ay.w);
  }

  // wave32 butterfly reduction (all lanes end with full sums)
#pragma unroll
  for (int m = 16; m >= 1; m >>= 1) {
    s1 += __shfl_xor(s1, m, 32);
    s2 += __shfl_xor(s2, m, 32);
    s3 += __shfl_xor(s3, m, 32);
  }

  const float inv  = 1.f / (float)H_FEAT;
  const float mu   = s1 * inv;
  float       var  = s2 * inv - mu * mu;
  var = fmaxf(var, 0.f);
  const float rstd = 1.f / sqrtf(var + LN_EPS);

  // delta = l0 - l1 ; softmax over 2 => w0 = sigmoid(delta)
  const float A  = scal[0];
  const float Cp = scal[1];
  const float delta = rstd * (s3 - mu * A) + Cp;
  const float w0 = 1.f / (1.f + __expf(-delta));
  const float w1 = 1.f - w0;

  float4* __restrict__ o4 = (float4*)out + (size_t)row * CH;
#pragma unroll
  for (int j = 0; j < 6; ++j) {
    const int c = lane + 32 * j;
    float4 o;
    o.x = w0 * sv[j].x + w1 * mv[j].x;
    o.y = w0 * sv[j].y + w1 * mv[j].y;
    o.z = w0 * sv[j].z + w1 * mv[j].z;
    o.w = w0 * sv[j].w + w1 * mv[j].w;
    o4[c] = o;
  }
}

// ---- Prep: a[i] = gamma[i]*(w0[i]-w1[i]); A = sum a; C' = sum beta*dw + (b0-b1)
__global__ __launch_bounds__(256) void ln_gate_prep(
    const float* __restrict__ gamma, const float* __restrict__ beta,
    const float* __restrict__ gw,    const float* __restrict__ gb,
    float* __restrict__ a_out, float* __restrict__ scal_out) {
  __shared__ float rA[256];
  __shared__ float rC[256];
  const int t = threadIdx.x;
  float accA = 0.f, accC = 0.f;
#pragma unroll
  for (int j = 0; j < H_FEAT / 256; ++j) {
    const int i = t + 256 * j;
    const float dw = gw[i] - gw[H_FEAT + i];
    const float ai = gamma[i] * dw;
    a_out[i] = ai;
    accA += ai;
    accC += beta[i] * dw;
  }
  rA[t] = accA; rC[t] = accC;
  __syncthreads();
  for (int s = 128; s > 0; s >>= 1) {
    if (t < s) { rA[t] += rA[t + s]; rC[t] += rC[t + s]; }
    __syncthreads();
  }
  if (t == 0) {
    scal_out[0] = rA[0];
    scal_out[1] = rC[0] + (gb[0] - gb[1]);
  }
}

extern "C" void kernel_launch(void* const* d_in, const int* in_sizes, int n_in,
                              void* d_out, int out_size, void* d_ws, size_t ws_size,
                              hipStream_t stream) {
  const float* seq   = (const float*)d_in[0];
  const float* msa   = (const float*)d_in[1];
  const float* gamma = (const float*)d_in[2];
  const float* beta  = (const float*)d_in[3];
  const float* gw    = (const float*)d_in[4];
  const float* gb    = (const float*)d_in[5];
  float* out = (float*)d_out;

  float* a_vec = (float*)d_ws;          // H_FEAT floats
  float* scal  = a_vec + H_FEAT;        // 2 floats: A, C'

  const int nrows  = in_sizes[0] / D_FEAT;            // 16384
  const int blocks = (nrows + ROWS_PER_BLK - 1) / ROWS_PER_BLK;

  ln_gate_prep<<<1, 256, 0, stream>>>(gamma, beta, gw, gb, a_vec, scal);
  ln_gate_fuse<<<blocks, THREADS, 0, stream>>>(seq, msa, a_vec, scal, out, nrows);
}